// DynamicTopKPool_69784628625744
// MI455X (gfx1250) — compile-verified
//
#include <hip/hip_runtime.h>
#include <hip/hip_bf16.h>
#include <math.h>

// Problem constants (match the reference).
#define B_GR    16
#define N_NODES 2048
#define F_DIM   128
#define K_POOL  1024

typedef __attribute__((ext_vector_type(2))) float v2f;
typedef __attribute__((ext_vector_type(8))) float v8f;

// ---------------------------------------------------------------------------
// Kernel 1: score[b,n] = tanh((x[b,n,:] . w) / ||w||)
// One wave computes 16 node scores using V_WMMA_F32_16X16X4_F32:
//   A tile = 16 nodes x 4 features (f32, exact), B tile = w chunk replicated
//   into all 16 N-columns, so D[m, :] = partial score of node m (broadcast).
// A 16x4 f32 layout: lanes 0-15 hold K=0 (v0) / K=1 (v1); lanes 16-31 K=2/K=3.
// B 4x16 f32 layout mirrors A (rows striped across lane halves), and since we
// replicate w across columns, b.x/b.y use the same k-offset as a.x/a.y.
// D 16x16 f32 layout: VGPR r = M=r (lanes 0-15) / M=8+r (lanes 16-31).
// ---------------------------------------------------------------------------
__global__ __launch_bounds__(256)
void dtkp_score_kernel(const float* __restrict__ x,
                       const float* __restrict__ w,
                       float* __restrict__ scores) {
    __shared__ float s_w[F_DIM];
    __shared__ float s_invnorm;

    const int tid = threadIdx.x;
    if (tid < F_DIM) s_w[tid] = w[tid];
    __syncthreads();
    if (tid == 0) {
        float acc = 0.0f;
        for (int i = 0; i < F_DIM; ++i) acc += s_w[i] * s_w[i];
        s_invnorm = rsqrtf(acc);   // 1 / ||w||
    }
    __syncthreads();
    const float invn = s_invnorm;

    const int lane      = tid & 31;
    const int wave      = tid >> 5;
    const int node_base = (blockIdx.x * 8 + wave) * 16;   // 16 nodes per wave
    const int m         = lane & 15;                      // node within tile
    const int ksel      = (lane >> 4) << 1;               // +0 (lanes 0-15) / +2 (16-31)

    const float* __restrict__ row = x + (size_t)(node_base + m) * F_DIM;

    v8f acc = {};
    #pragma unroll 4
    for (int k0 = 0; k0 < F_DIM; k0 += 4) {
        const int ka = k0 + ksel;
        v2f a; a.x = row[ka];  a.y = row[ka + 1];
        v2f b; b.x = s_w[ka];  b.y = s_w[ka + 1];
        __builtin_prefetch(row + ka + 32, 0, 0);  // global_prefetch_b8, speculative
        // 8 args: (neg_a, A, neg_b, B, c_mod, C, reuse_a, reuse_b)
        acc = __builtin_amdgcn_wmma_f32_16x16x4_f32(
            false, a, false, b, (short)0, acc, false, false);
    }

    // All N-columns of D are identical; extract per-row scores from lane 0
    // (rows M=0..7) and lane 16 (rows M=8..15).
    if (lane == 0) {
        #pragma unroll
        for (int r = 0; r < 8; ++r)
            scores[node_base + r] = tanhf(acc[r] * invn);
    } else if (lane == 16) {
        #pragma unroll
        for (int r = 0; r < 8; ++r)
            scores[node_base + 8 + r] = tanhf(acc[r] * invn);
    }
}

// ---------------------------------------------------------------------------
// Kernel 2: per-graph full bitonic sort of 2048 (score, idx) pairs in LDS,
// descending by score, ties -> lower original index (jax.lax.top_k order).
// One block (1024 threads) per graph; emits the top K_POOL = 1024 entries.
// ---------------------------------------------------------------------------
__global__ __launch_bounds__(1024)
void dtkp_topk_kernel(const float* __restrict__ scores,
                      float* __restrict__ vals,
                      int* __restrict__ idxs) {
    __shared__ float s_s[N_NODES];
    __shared__ int   s_i[N_NODES];

    const int b = blockIdx.x;
    for (int i = threadIdx.x; i < N_NODES; i += 1024) {
        s_s[i] = scores[b * N_NODES + i];
        s_i[i] = i;
    }
    __syncthreads();

    for (unsigned k = 2; k <= N_NODES; k <<= 1) {
        for (unsigned j = k >> 1; j > 0; j >>= 1) {
            for (unsigned i = threadIdx.x; i < N_NODES; i += 1024) {
                const unsigned p = i ^ j;
                if (p > i) {
                    const float si = s_s[i], sp = s_s[p];
                    const int   di = s_i[i], dp = s_i[p];
                    // "i belongs before p" under descending-with-index-tiebreak
                    const bool i_first = (si > sp) || (si == sp && di < dp);
                    const bool desc    = ((i & k) == 0);
                    if (desc ? !i_first : i_first) {
                        s_s[i] = sp; s_s[p] = si;
                        s_i[i] = dp; s_i[p] = di;
                    }
                }
            }
            __syncthreads();
        }
    }

    // Top K_POOL == blockDim: one element per thread.
    vals[b * K_POOL + threadIdx.x] = s_s[threadIdx.x];
    idxs[b * K_POOL + threadIdx.x] = s_i[threadIdx.x];
}

// ---------------------------------------------------------------------------
// Kernel 3: out[row, :] = x[b*N + idx[row], :] * vals[row]  (row-major, F=128)
// One wave per output row: 32 lanes x float4 = 128 floats, coalesced 16B ops.
// Also writes new_batch (graph id per kept node) after the feature block.
// ---------------------------------------------------------------------------
__global__ __launch_bounds__(256)
void dtkp_gather_kernel(const float* __restrict__ x,
                        const float* __restrict__ vals,
                        const int* __restrict__ idxs,
                        float* __restrict__ out,
                        int out_size) {
    const int lane = threadIdx.x & 31;
    const int row  = blockIdx.x * 8 + (threadIdx.x >> 5);   // 8 rows per block
    const int b    = row >> 10;                             // row / K_POOL
    const int src  = b * N_NODES + idxs[row];
    const float v  = vals[row];

    const float4* __restrict__ srcp = (const float4*)(x + (size_t)src * F_DIM);
    float4* __restrict__ dstp       = (float4*)(out + (size_t)row * F_DIM);

    float4 d = srcp[lane];
    d.x *= v; d.y *= v; d.z *= v; d.w *= v;
    dstp[lane] = d;

    if (lane == 0) {
        const size_t boff = (size_t)B_GR * K_POOL * F_DIM;
        if ((size_t)out_size >= boff + (size_t)B_GR * K_POOL)
            out[boff + row] = (float)b;   // new_batch
    }
}

// ---------------------------------------------------------------------------
extern "C" void kernel_launch(void* const* d_in, const int* in_sizes, int n_in,
                              void* d_out, int out_size, void* d_ws, size_t ws_size,
                              hipStream_t stream) {
    (void)in_sizes; (void)n_in; (void)ws_size;
    const float* x = (const float*)d_in[0];        // [B*N, F] f32
    const float* w = (const float*)d_in[1];        // [F] f32
    // d_in[2] (batch, int64) is unused: new_batch is just repeat(arange(B)).

    float* scores = (float*)d_ws;                          // B*N      floats
    float* vals   = scores + B_GR * N_NODES;               // B*K_POOL floats
    int*   idxs   = (int*)(vals + B_GR * K_POOL);          // B*K_POOL ints
    float* out    = (float*)d_out;

    // 1) scores: 2048 waves total, 8 waves per 256-thread block -> 256 blocks
    dtkp_score_kernel<<<(B_GR * N_NODES) / (16 * 8), 256, 0, stream>>>(x, w, scores);

    // 2) per-graph bitonic top-k: one block per graph
    dtkp_topk_kernel<<<B_GR, 1024, 0, stream>>>(scores, vals, idxs);

    // 3) gather + scale + new_batch: 8 rows per block
    dtkp_gather_kernel<<<(B_GR * K_POOL) / 8, 256, 0, stream>>>(x, vals, idxs, out, out_size);
}